// Conv2d_37546604102083
// MI455X (gfx1250) — compile-verified
//
#include <hip/hip_runtime.h>

typedef __attribute__((ext_vector_type(2))) float v2f;
typedef __attribute__((ext_vector_type(8))) float v8f;

#define BATCH 32
#define HIN   224
#define WIN   224
#define KCH   64
#define KS    7
#define OHW   218      // 224 - 7 + 1
#define TAPS  49
#define KPAD  64       // 8 kr-rows x 8 taps (kr=7 row unused), power-of-2 addressing
#define NSTEP 14       // 7 kernel rows x 2 steps (4 taps each, 8th tap zero-weight)
#define ROWS  4        // output rows per block
#define INW   40       // staged input width: 32 + 6, padded to 40
#define INH   (ROWS + KS - 1)   // 10 staged input rows

__global__ __launch_bounds__(256) void conv2d_wmma_f32(
    const float* __restrict__ x,        // [32][224][224]
    const float* __restrict__ kernels,  // [64][7][7]
    float* __restrict__ out)            // [32][64][218][218]
{
    __shared__ float sW[KCH * KPAD];    // [ch][8*kr + i], i==7 or kr==7 -> 0
    __shared__ float sIn[INH * INW];    // [row][col] input window

    const int tid = threadIdx.x;
    const int ow0 = blockIdx.x * 32;    // 7 blocks cover 218 (last partial)
    const int oh0 = blockIdx.y * ROWS;  // 55 blocks cover 218 (last partial)
    const int b   = blockIdx.z;

    // ---- stage weights: sW[ch][8*kr + i] (shift/mask-only indexing) ----
    for (int i = tid; i < KCH * KPAD; i += 256) {
        const int ch = i >> 6;
        const int tp = i & 63;
        const int kr = tp >> 3;
        const int ic = tp & 7;
        sW[i] = (kr < KS && ic < KS) ? kernels[ch * TAPS + kr * KS + ic] : 0.0f;
    }
    // ---- stage input window: rows oh0..oh0+9, cols ow0..ow0+39 (clamped) ----
    for (int i = tid; i < INH * INW; i += 256) {
        const int r = i / INW;
        const int c = i % INW;
        int ih = oh0 + r; if (ih > HIN - 1) ih = HIN - 1;
        int iw = ow0 + c; if (iw > WIN - 1) iw = WIN - 1;
        sIn[i] = x[(b * HIN + ih) * WIN + iw];
    }
    __syncthreads();

    const int lane   = tid & 31;
    const int wave   = tid >> 5;
    const int lo     = lane & 15;       // ow within tile / ch within tile
    const int half   = lane >> 4;       // selects K sub-pair {0,1} vs {2,3}
    const int owTile = (wave & 1) * 16; // 2 pixel tiles
    const int chTile = (wave >> 1) * 16;// 4 channel tiles

    // half-dependence folded into base pointers ONCE; all per-step offsets
    // below are compile-time immediates.
    const float* sWB  = &sW[((chTile + lo) << 6) + 2 * half];
    const float* sInB = &sIn[owTile + lo + 2 * half];

    // ---- A fragments (weights): step s -> taps 4s, 4s+1 (+2*half in base) ----
    v2f afrag[NSTEP];
#pragma unroll
    for (int s = 0; s < NSTEP; ++s) {
        afrag[s].x = sWB[4 * s];
        afrag[s].y = sWB[4 * s + 1];
    }

    const int ow = ow0 + owTile + lo;
#pragma unroll
    for (int r = 0; r < ROWS; ++r) {
        v8f acc = {};
#pragma unroll
        for (int s = 0; s < NSTEP; ++s) {
            // step s = 2t+(0|1): kernel row t, tap group base t*INW + 0|4
            const int ofs = (s >> 1) * INW + ((s & 1) ? 4 : 0) + r * INW;
            v2f bfrag;
            bfrag.x = sInB[ofs];        // immediate DS offsets, adjacent pair
            bfrag.y = sInB[ofs + 1];
            // D = A(16ch x 4tap) * B(4tap x 16ow) + C
            acc = __builtin_amdgcn_wmma_f32_16x16x4_f32(
                false, afrag[s], false, bfrag, (short)0, acc, false, false);
        }
        const int oh = oh0 + r;
        if (oh < OHW && ow < OHW) {
            // base address for ch = chTile + 8*half; per-v stores use
            // immediate offsets (v * 218*218 floats, < 24-bit offset range)
            float* outB = &out[(((size_t)b * KCH + chTile + 8 * half) * OHW + oh) * OHW + ow];
#pragma unroll
            for (int v = 0; v < 8; ++v) {
                outB[(size_t)v * OHW * OHW] = acc[v];
            }
        }
    }
}

extern "C" void kernel_launch(void* const* d_in, const int* in_sizes, int n_in,
                              void* d_out, int out_size, void* d_ws, size_t ws_size,
                              hipStream_t stream) {
    const float* x       = (const float*)d_in[0];   // 32*224*224
    const float* kernels = (const float*)d_in[1];   // 64*7*7
    float* out           = (float*)d_out;           // 32*64*218*218

    dim3 grid((OHW + 31) / 32,          // 7 ow-blocks
              (OHW + ROWS - 1) / ROWS,  // 55 oh-blocks
              BATCH);                   // 32 batches
    dim3 block(256);
    conv2d_wmma_f32<<<grid, block, 0, stream>>>(x, kernels, out);
}